// LQEPose_19988777796195
// MI455X (gfx1250) — compile-verified
//
#include <hip/hip_runtime.h>
#include <hip/hip_bf16.h>
#include <float.h>

#define TOPK    4
#define NUM_KPT 17
#define HID     64
#define CCH     256
#define FH      160
#define FW      160
#define BATCH   4
#define NQ      1000
#define KPAD    88          // 85 (=17*5) padded to a multiple of 4 for K4 WMMA steps

typedef __attribute__((ext_vector_type(2))) float v2f;
typedef __attribute__((ext_vector_type(8))) float v8f;

// ---------------------------------------------------------------------------
// Kernel 1: bilinear sample + per-point top-4 + mean  -> stat matrix (4000 x 88)
// One wave32 per (b, l, kpt) sample point; lane L owns channels L, L+32, ...
// ---------------------------------------------------------------------------
__global__ __launch_bounds__(256) void lqe_sample_topk(
    const float* __restrict__ pred,   // (B, NQ, 2*NUM_KPT)
    const float* __restrict__ feat,   // (B, C, H, W)
    float* __restrict__ xp)           // (B*NQ, KPAD) stat output
{
    const int wave = (blockIdx.x * blockDim.x + threadIdx.x) >> 5;
    const int lane = threadIdx.x & 31;
    const int P = BATCH * NQ * NUM_KPT;
    if (wave >= P) return;

    const int b   = wave / (NQ * NUM_KPT);
    const int rem = wave - b * (NQ * NUM_KPT);
    const int l   = rem / NUM_KPT;
    const int k   = rem - l * NUM_KPT;
    const int row = b * NQ + l;

    // grid coords -> pixel coords (align_corners=False style, per reference)
    const float px = pred[row * 2 * NUM_KPT + 2 * k];
    const float py = pred[row * 2 * NUM_KPT + 2 * k + 1];
    const float gx = 2.0f * px - 1.0f;
    const float gy = 2.0f * py - 1.0f;
    const float ix = ((gx + 1.0f) * (float)FW - 1.0f) * 0.5f;
    const float iy = ((gy + 1.0f) * (float)FH - 1.0f) * 0.5f;
    const float x0f = floorf(ix), y0f = floorf(iy);
    const float wx1 = ix - x0f, wx0 = 1.0f - wx1;
    const float wy1 = iy - y0f, wy0 = 1.0f - wy1;
    const int x0 = (int)x0f, y0 = (int)y0f;
    const int x1 = x0 + 1,   y1 = y0 + 1;

    const bool vx0 = (x0 >= 0) && (x0 < FW);
    const bool vx1 = (x1 >= 0) && (x1 < FW);
    const bool vy0 = (y0 >= 0) && (y0 < FH);
    const bool vy1 = (y1 >= 0) && (y1 < FH);
    const int xc0 = min(max(x0, 0), FW - 1), xc1 = min(max(x1, 0), FW - 1);
    const int yc0 = min(max(y0, 0), FH - 1), yc1 = min(max(y1, 0), FH - 1);

    // fold validity masks into the bilinear weights (== reference's gather*valid)
    const float e00 = (vx0 && vy0) ? (wx0 * wy0) : 0.0f;
    const float e10 = (vx1 && vy0) ? (wx1 * wy0) : 0.0f;
    const float e01 = (vx0 && vy1) ? (wx0 * wy1) : 0.0f;
    const float e11 = (vx1 && vy1) ? (wx1 * wy1) : 0.0f;

    const int hw = FH * FW;
    const float* __restrict__ fb = feat + (long)b * CCH * hw;
    const int o00 = yc0 * FW + xc0, o10 = yc0 * FW + xc1;
    const int o01 = yc1 * FW + xc0, o11 = yc1 * FW + xc1;

    // per-lane top-4, sorted descending
    float t0 = -FLT_MAX, t1 = -FLT_MAX, t2 = -FLT_MAX, t3 = -FLT_MAX;
    #pragma unroll
    for (int i = 0; i < CCH / 32; ++i) {
        const int c = lane + 32 * i;
        const float* __restrict__ p = fb + (long)c * hw;
        float v = p[o00] * e00 + p[o10] * e10 + p[o01] * e01 + p[o11] * e11;
        // insert v into sorted chain (push out the min)
        float m;
        m = fmaxf(t0, v); v = fminf(t0, v); t0 = m;
        m = fmaxf(t1, v); v = fminf(t1, v); t1 = m;
        m = fmaxf(t2, v); v = fminf(t2, v); t2 = m;
        m = fmaxf(t3, v); v = fminf(t3, v); t3 = m;
    }

    // butterfly: merge sorted descending 4-lists across the wave32
    #pragma unroll
    for (int s = 16; s >= 1; s >>= 1) {
        float ta[4] = { t0, t1, t2, t3 };
        float oa[4] = { __shfl_xor(t0, s, 32), __shfl_xor(t1, s, 32),
                        __shfl_xor(t2, s, 32), __shfl_xor(t3, s, 32) };
        float ma[4];
        int i2 = 0, j2 = 0;
        #pragma unroll
        for (int n = 0; n < 4; ++n) {
            const bool takeT = (j2 >= 4) || ((i2 < 4) && (ta[i2] >= oa[j2]));
            ma[n] = takeT ? ta[i2] : oa[j2];
            if (takeT) ++i2; else ++j2;
        }
        t0 = ma[0]; t1 = ma[1]; t2 = ma[2]; t3 = ma[3];
    }

    if (lane == 0) {
        float* xr = xp + (long)row * KPAD + k * (TOPK + 1);
        xr[0] = t0; xr[1] = t1; xr[2] = t2; xr[3] = t3;
        xr[4] = 0.25f * (t0 + t1 + t2 + t3);
        if (k == NUM_KPT - 1) {             // zero K-padding columns 85..87
            float* pr = xp + (long)row * KPAD;
            pr[85] = 0.0f; pr[86] = 0.0f; pr[87] = 0.0f;
        }
    }
}

// ---------------------------------------------------------------------------
// Kernel 2: MLP head with fp32 WMMA.
//   h   = relu( x(4000x85) @ W1(85x64) + b1 )      via v_wmma_f32_16x16x4_f32
//   out = scores + h @ W2(64x1) + b2               via LDS + 64-wide dot
// Block = 128 threads (4 waves); block handles 16 rows; wave w handles N-tile w.
// ---------------------------------------------------------------------------
__global__ __launch_bounds__(128) void lqe_mlp(
    const float* __restrict__ xp,     // (4000, KPAD)
    const float* __restrict__ W1,     // (85, 64) row-major
    const float* __restrict__ b1,     // (64)
    const float* __restrict__ W2,     // (64)
    const float* __restrict__ b2,     // (1)
    const float* __restrict__ scores, // (4000)
    float* __restrict__ out)          // (4000)
{
    __shared__ float hbuf[16][HID];

    const int tid  = threadIdx.x;
    const int lane = tid & 31;
    const int w    = tid >> 5;               // wave id -> N tile [16w, 16w+16)
    const int row0 = blockIdx.x * 16;

    // A frag (16x4 f32, ISA 7.12.2): lane = M in [0,16); half-wave selects K pair
    const int m     = lane & 15;
    const int khalf = (lane >> 4) << 1;      // 0 for lanes 0-15, 2 for lanes 16-31
    const int n     = w * 16 + m;            // B/C/D: lane%16 = N column

    const float* __restrict__ xr = xp + (long)row0 * KPAD;

    v8f acc = {};
    #pragma unroll
    for (int kk = 0; kk < KPAD / 4; ++kk) {
        const int kb = kk * 4 + khalf;
        v2f a;                               // A: x[row0+m][kb], x[row0+m][kb+1]
        a.x = xr[m * KPAD + kb];
        a.y = xr[m * KPAD + kb + 1];
        v2f bf;                              // B: W1[kb][n], W1[kb+1][n] (zero-pad K>=85)
        bf.x = (kb     < 85) ? W1[kb * HID + n]       : 0.0f;
        bf.y = (kb + 1 < 85) ? W1[(kb + 1) * HID + n] : 0.0f;
        acc = __builtin_amdgcn_wmma_f32_16x16x4_f32(
            /*neg_a=*/false, a, /*neg_b=*/false, bf,
            /*c_mod=*/(short)0, acc, /*reuse_a=*/false, /*reuse_b=*/false);
    }

    // C/D layout: VGPR r -> M = r + 8*(lane>=16), N = lane%16 (+16w)
    const int mbase = (lane >> 4) * 8;
    const float bias = b1[n];
    #pragma unroll
    for (int r = 0; r < 8; ++r) {
        hbuf[mbase + r][n] = fmaxf(acc[r] + bias, 0.0f);
    }
    __syncthreads();

    if (tid < 16) {
        float s = b2[0];
        #pragma unroll
        for (int c = 0; c < HID; ++c) s += hbuf[tid][c] * W2[c];
        const int gi = row0 + tid;
        out[gi] = scores[gi] + s;
    }
}

// ---------------------------------------------------------------------------
extern "C" void kernel_launch(void* const* d_in, const int* in_sizes, int n_in,
                              void* d_out, int out_size, void* d_ws, size_t ws_size,
                              hipStream_t stream) {
    (void)in_sizes; (void)n_in; (void)out_size; (void)ws_size;
    const float* scores = (const float*)d_in[0];
    const float* pred   = (const float*)d_in[1];
    const float* feat   = (const float*)d_in[2];
    const float* W1     = (const float*)d_in[3];
    const float* b1     = (const float*)d_in[4];
    const float* W2     = (const float*)d_in[5];
    const float* b2     = (const float*)d_in[6];
    float* out = (float*)d_out;
    float* xp  = (float*)d_ws;               // (4000 x 88) f32 = 1.41 MB scratch

    const int P = BATCH * NQ * NUM_KPT;      // 68000 sample points = 68000 waves
    const int blocks1 = (P * 32 + 255) / 256;
    lqe_sample_topk<<<blocks1, 256, 0, stream>>>(pred, feat, xp);

    lqe_mlp<<<(BATCH * NQ) / 16, 128, 0, stream>>>(xp, W1, b1, W2, b2, scores, out);
}